// RGWRP_63367947485762
// MI455X (gfx1250) — compile-verified
//
#include <hip/hip_runtime.h>
#include <hip/hip_bf16.h>

#define HW       4096
#define K_TOP    905
#define THREADS  256

typedef __attribute__((ext_vector_type(2))) float v2f;
typedef __attribute__((ext_vector_type(8))) float v8f;

// ---- CDNA5 async global -> LDS copy (ASYNCcnt path, inline asm) ------------
// CDNA5 ISA: GLOBAL_LOAD_ASYNC_TO_LDS_B128, GV mode:
//   VDST = VGPR holding LDS byte address, VADDR = VGPR pair w/ 64-bit address.
// Flat address of a __shared__ object: addr[31:0] == LDS byte offset (aperture
// layout, CDNA5 ISA), so truncating the generic pointer is valid.
__device__ __forceinline__ void async_copy_b128(const void* gsrc, void* lds_dst) {
  unsigned           laddr = (unsigned)(unsigned long long)lds_dst;
  unsigned long long gaddr = (unsigned long long)gsrc;
  asm volatile("global_load_async_to_lds_b128 %0, %1, off"
               :: "v"(laddr), "v"(gaddr)
               : "memory");
}

__device__ __forceinline__ void wait_async_zero() {
#if defined(__gfx1250__) && __has_builtin(__builtin_amdgcn_s_wait_asynccnt)
  __builtin_amdgcn_s_wait_asynccnt(0);
#else
  asm volatile("s_wait_asynccnt 0" ::: "memory");
#endif
}

// ---- kernel: one workgroup per (b,c) row -----------------------------------
__global__ __launch_bounds__(THREADS) void
RGWRP_gwrp_topk_kernel(const float* __restrict__ x, float* __restrict__ out) {
  __shared__ unsigned keys[HW];       // 16 KB: row data, then monotonic keys
  __shared__ float    red[THREADS];   // 1 KB : per-thread partial sums

  const int tid = threadIdx.x;
  const long long row = blockIdx.x;
  const float* src = x + row * (long long)HW;

  // 1) Async-DMA the 16KB row into LDS: 4 x b128 per lane, contiguous 16B/lane.
  {
    const char* g = (const char*)src;
    char*       l = (char*)keys;
#pragma unroll
    for (int s = 0; s < 4; ++s) {
      const int off = s * 4096 + tid * 16;
      async_copy_b128(g + off, l + off);
    }
    wait_async_zero();   // each wave waits on its own ASYNCcnt
  }
  __syncthreads();       // cross-wave visibility of LDS

  // 2) fp32 -> order-preserving u32 key (descending float == descending key).
#pragma unroll
  for (int i = tid; i < HW; i += THREADS) {
    const unsigned u = keys[i];
    keys[i] = (u & 0x80000000u) ? ~u : (u | 0x80000000u);
  }
  __syncthreads();

  // 3) Bitonic sort, descending, 4096 keys, 256 threads (8 CE/thread/pass).
  for (unsigned k = 2; k <= HW; k <<= 1) {
    for (unsigned j = k >> 1; j > 0; j >>= 1) {
      for (unsigned i = tid; i < HW; i += THREADS) {
        const unsigned ixj = i ^ j;
        if (ixj > i) {
          const unsigned a = keys[i];
          const unsigned b = keys[ixj];
          const bool desc = ((i & k) == 0);
          if (desc ? (a < b) : (a > b)) { keys[i] = b; keys[ixj] = a; }
        }
      }
      __syncthreads();
    }
  }

  // 4) Weighted partial sums over the sorted top-905.
  //    w[i] = decay^i = 2^(i * log2(0.01)/904)
  const float c1 = -0.0073493984f;    // log2(0.01) / 904
  float part = 0.0f;
  for (int i = tid; i < K_TOP; i += THREADS) {
    const unsigned key = keys[i];
    const unsigned u = (key & 0x80000000u) ? (key ^ 0x80000000u) : ~key;
    const float v = __uint_as_float(u);
    part += v * __builtin_exp2f((float)i * c1);
  }
  red[tid] = part;
  __syncthreads();

  // 5) Wave 0: reduce 256 partials with 4 chained v_wmma_f32_16x16x4_f32.
  //    B = ones(4x16) (layout-invariant), A chunks cover red[0..255].
  //    D[i][j] = sum_k sum_chunk A[i][k]  => every column holds the 16 rowsums.
  if (tid < 32) {
    v2f ones; ones.x = 1.0f; ones.y = 1.0f;
    v8f acc = {};
    // A layout (32-bit 16x4): lanes 0-15 -> (row=lane, K0/K1); lanes 16-31 -> (row=lane-16, K2/K3)
    const int base = (tid < 16) ? (tid * 4) : ((tid - 16) * 4 + 2);
#pragma unroll
    for (int chunk = 0; chunk < 4; ++chunk) {
      v2f a;
      a.x = red[chunk * 64 + base];
      a.y = red[chunk * 64 + base + 1];
      acc = __builtin_amdgcn_wmma_f32_16x16x4_f32(
          /*neg_a=*/false, a, /*neg_b=*/false, ones,
          /*c_mod=*/(short)0, acc, /*reuse_a=*/false, /*reuse_b=*/false);
    }
    // lane 0 holds D[0..7][0], lane 16 holds D[8..15][0]
    float s = acc[0] + acc[1] + acc[2] + acc[3] + acc[4] + acc[5] + acc[6] + acc[7];
    s += __shfl_xor(s, 16, 32);
    if (tid == 0) {
      // SUM_W = (1 - d^905)/(1 - d) with d = 0.01^(1/904); d^905 = 0.01*d
      const double d = exp(log(0.01) / 904.0);
      const double sumw = (1.0 - 0.01 * d) / (1.0 - d);
      out[row] = (float)((double)s / sumw);
    }
  }
}

extern "C" void kernel_launch(void* const* d_in, const int* in_sizes, int n_in,
                              void* d_out, int out_size, void* d_ws, size_t ws_size,
                              hipStream_t stream) {
  const float* x = (const float*)d_in[0];
  float* out = (float*)d_out;
  const int rows = in_sizes[0] / HW;   // 64*256 = 16384
  RGWRP_gwrp_topk_kernel<<<dim3(rows), dim3(THREADS), 0, stream>>>(x, out);
}